// kernel_generated_4_21887153341170
// MI455X (gfx1250) — compile-verified
//
#include <hip/hip_runtime.h>
#include <hip/hip_bf16.h>

typedef __attribute__((ext_vector_type(2))) float v2f;
typedef __attribute__((ext_vector_type(8))) float v8f;

// ---------------------------------------------------------------------------
// Kernel 1: fuse w1 (128,9) and w2 (32,7,9) into Wc with K-order r = kk*32+j:
//   Wc[oc*224 + kk*32 + j] = sum_i w1[oc*9+i] * w2[(j*7+kk)*9+i]
// This K permutation makes the GEMM inner loop free of div/mod: within any
// aligned group of 4 consecutive K indices the tap kk is constant and the
// channel j advances by 1 (B-address stride = 784 elements).
// ---------------------------------------------------------------------------
__global__ void fuse_weights_kernel(const float* __restrict__ w1,
                                    const float* __restrict__ w2,
                                    float* __restrict__ Wc) {
    int t = blockIdx.x * blockDim.x + threadIdx.x;
    if (t >= 128 * 224) return;
    int oc = t / 224;
    int r  = t - oc * 224;
    int kk = r >> 5;          // tap 0..6
    int j  = r & 31;          // channel 0..31
    float s = 0.0f;
#pragma unroll
    for (int i = 0; i < 9; ++i)
        s += w1[oc * 9 + i] * w2[(j * 7 + kk) * 9 + i];
    Wc[t] = s;
}

// ---------------------------------------------------------------------------
// Kernel 2: grouped conv as WMMA f32 GEMM (M=128 oc, K=224 (kk,j), N=784 hw).
//   Per group g: O[oc, p] = sum_{kk,j} Wc[oc, kk*32+j] * x[g*32+j, h, w+kk-3]
//   One wave = one 16x16 tile via V_WMMA_F32_16X16X4_F32, 7*8 = 56 K-steps.
//   Block = 256 threads = 8 waves = the 8 M-tiles; grid = 2 groups * 49 N-tiles.
//   Store applies roll(+1) along h.
// ---------------------------------------------------------------------------
__global__ __launch_bounds__(256) void conv_wmma_kernel(
    const float* __restrict__ x,     // (64, 28, 28)
    const float* __restrict__ Wc,    // (128, 224) permuted K
    float* __restrict__ out) {       // (256, 28, 28)
    const int blk   = blockIdx.x;
    const int g     = blk / 49;            // group 0..1
    const int nTile = blk % 49;            // N tile 0..48
    const int wave  = threadIdx.x >> 5;    // M tile 0..7
    const int lane  = threadIdx.x & 31;
    const int n     = lane & 15;           // tile column (N) / tile row (M) id
    const int half  = lane >> 4;           // lane half selects K pair

    const int mBase = wave * 16;
    const int p     = nTile * 16 + n;      // spatial position 0..783
    const int h     = p / 28;
    const int w     = p - h * 28;

    // A row pointer: this lane always reads row M = mBase + n of Wc.
    const float* __restrict__ wcRow = Wc + (mBase + n) * 224 + half * 2;
    // B base: group slice, row h; channel j strides by 784.
    const float* __restrict__ xrow  = x + g * (32 * 784) + h * 28;

    v8f acc = {};

#pragma unroll
    for (int kk = 0; kk < 7; ++kk) {
        const int  wx    = w + kk - 3;
        const bool valid = (wx >= 0) && (wx < 28);
        const float* __restrict__ xcol = xrow + wx + half * 2 * 784;
        const float* __restrict__ wcK  = wcRow + kk * 32;

#pragma unroll
        for (int jb = 0; jb < 32; jb += 4) {
            // A fragment: aligned 8-byte load (K = kk*32 + jb + half*2 + v)
            v2f a = *(const v2f*)(wcK + jb);

            // B fragment: constant-tap gather, stride 784 per channel
            v2f b;
            b.x = valid ? xcol[(jb + 0) * 784] : 0.0f;
            b.y = valid ? xcol[(jb + 1) * 784] : 0.0f;

            acc = __builtin_amdgcn_wmma_f32_16x16x4_f32(
                /*neg_a=*/false, a, /*neg_b=*/false, b,
                /*c_mod=*/(short)0, acc, /*reuse_a=*/false, /*reuse_b=*/false);
        }
    }

    // Store with roll(+1) along h. D layout: VGPR v holds row M = half*8 + v.
    int hRoll = h + 1;
    if (hRoll == 28) hRoll = 0;
    float* __restrict__ obase =
        out + ((g * 128 + mBase + half * 8) * 28 + hRoll) * 28 + w;
#pragma unroll
    for (int v = 0; v < 8; ++v)
        obase[v * 784] = acc[v];
}

// ---------------------------------------------------------------------------
extern "C" void kernel_launch(void* const* d_in, const int* in_sizes, int n_in,
                              void* d_out, int out_size, void* d_ws, size_t ws_size,
                              hipStream_t stream) {
    const float* x  = (const float*)d_in[0];   // 64*28*28
    const float* w1 = (const float*)d_in[1];   // 128*9
    const float* w2 = (const float*)d_in[2];   // 32*7*9
    float* out = (float*)d_out;                // 256*28*28
    float* Wc  = (float*)d_ws;                 // 128*224 floats scratch

    fuse_weights_kernel<<<(128 * 224 + 255) / 256, 256, 0, stream>>>(w1, w2, Wc);
    conv_wmma_kernel<<<2 * 49, 256, 0, stream>>>(x, Wc, out);
}